// Encoder_50826642980972
// MI455X (gfx1250) — compile-verified
//
#include <hip/hip_runtime.h>

typedef __attribute__((ext_vector_type(2))) float v2f;
typedef __attribute__((ext_vector_type(8))) float v8f;
typedef const float __attribute__((address_space(1)))* gptr; // force global_load path

#define D 256
#define NCOL 39
#define RPB 16        // rows per block
#define THREADS 512   // 16 waves (wave32)

__global__ __launch_bounds__(THREADS)
void encoder_wmma_kernel(const int*   __restrict__ entity,
                         const float* __restrict__ Wsp, const float* __restrict__ bsp,
                         const float* __restrict__ Wab, const float* __restrict__ bab,
                         const float* __restrict__ Wit, const float* __restrict__ bitm,
                         const float* __restrict__ Wm1, const float* __restrict__ Wm2,
                         const float* __restrict__ bmv,
                         const float* __restrict__ Wlv, const float* __restrict__ blv,
                         const float* __restrict__ Whp, const float* __restrict__ bhp,
                         const float* __restrict__ Wvl, const float* __restrict__ bvl,
                         const float* __restrict__ Wft, const float* __restrict__ bft,
                         const float* __restrict__ Wbl, const float* __restrict__ bbl,
                         float* __restrict__ out, int N)
{
    __shared__ int   s_ent[RPB * NCOL];            // staged entity rows
    __shared__ float s_feat[RPB * 64];             // 16 x 64 feature matrix (f32, K-padded)
    __shared__ float s_emb[RPB * D];               // gather-phase partial sums
    __shared__ __align__(16) const float* s_wrow[64]; // concatenated-weight row pointers

    const int tid  = threadIdx.x;
    const int lane = tid & 31;
    const int wave = tid >> 5;
    const int row0 = blockIdx.x * RPB;

    // ---- stage entity rows into LDS + build weight-row pointer table ----
    for (int i = tid; i < RPB * NCOL; i += THREADS) {
        int r = i / NCOL, c = i - r * NCOL;
        int gr = row0 + r; if (gr >= N) gr = N - 1;   // clamp (N is multiple of 16 anyway)
        s_ent[i] = entity[(size_t)gr * NCOL + c];
    }
    if (tid < 64) {
        int k = tid;
        const float* p;
        if (k < 7)       p = Wlv + (size_t)k * D;
        else if (k < 17) p = Whp + (size_t)(k - 7) * D;
        else if (k < 53) p = Wvl + (size_t)(k - 17) * D;
        else if (k < 62) p = Wft + (size_t)(k - 53) * D;
        else             p = Wft;  // pad rows: paired features are 0, product is 0
        s_wrow[k] = p;
    }
    __syncthreads();

    // ---- build 16x64 feature matrix in LDS ----
    for (int i = tid; i < RPB * 64; i += THREADS) {
        int r = i >> 6, k = i & 63;
        const int* e = &s_ent[r * NCOL];
        float v;
        if (k < 7)        v = (float)((e[7] >> k) & 1);                         // level bits
        else if (k < 17)  v = (float)((e[6] >> (k - 7)) & 1);                   // hp bits
        else if (k < 53)  { int kk = k - 17;
                            v = (float)((e[30 + (kk >> 2)] >> (kk & 3)) & 1); } // vol bits
        else if (k == 53) v = (float)e[6] * (1.0f / 1023.0f);                   // feats
        else if (k == 54) v = (float)e[7] * (1.0f / 100.0f);
        else if (k < 62)  v = (float)e[23 + (k - 55)] * 0.5f;
        else              v = 0.0f;                                             // K pad
        s_feat[i] = v;
    }

    // ---- gather phase: wave w handles row w, lane handles 8 contiguous cols ----
    {
        const int* e = &s_ent[wave * NCOL];
        const int col0 = lane * 8;
        float4 a0 = {0.f, 0.f, 0.f, 0.f}, a1 = {0.f, 0.f, 0.f, 0.f};

        auto add2 = [&](const float* __restrict__ p) {
            const float4* q = (const float4*)(p + col0);
            float4 t0 = q[0], t1 = q[1];
            a0.x += t0.x; a0.y += t0.y; a0.z += t0.z; a0.w += t0.w;
            a1.x += t1.x; a1.y += t1.y; a1.z += t1.z; a1.w += t1.w;
        };
        auto fma2 = [&](float s, const float* __restrict__ p) {
            const float4* q = (const float4*)(p + col0);
            float4 t0 = q[0], t1 = q[1];
            a0.x += s * t0.x; a0.y += s * t0.y; a0.z += s * t0.z; a0.w += s * t0.w;
            a1.x += s * t1.x; a1.y += s * t1.y; a1.z += s * t1.z; a1.w += s * t1.w;
        };

        add2(Wsp + (size_t)e[0] * D);
        add2(Wab + (size_t)e[1] * D);
        add2(Wit + (size_t)e[2] * D);
        // all 9 bias vectors (added once per row)
        add2(bsp); add2(bab); add2(bitm); add2(bmv); add2(blv);
        add2(bhp); add2(bvl); add2(bft);  add2(bbl);

        #pragma unroll
        for (int m = 0; m < 4; ++m) {
            int   mi = e[8 + m];
            float pp = (float)e[12 + m] * (1.0f / 1023.0f);
            add2(Wm1 + (size_t)mi * D);
            fma2(pp, Wm2 + (size_t)mi * D);
        }

        const int bool_col[16] = {3,17,20,21,22,18,19,5,4,23,24,25,26,27,28,29};
        const int bool_add[16] = {0,3,10,12,14,16,24,28,30,38,51,64,77,90,103,116}; // SHIFT+OFF
        #pragma unroll
        for (int i2 = 0; i2 < 16; ++i2) {
            int bi = e[bool_col[i2]] + bool_add[i2];
            add2(Wbl + (size_t)bi * D);
        }

        *(float4*)&s_emb[wave * D + col0]     = a0;
        *(float4*)&s_emb[wave * D + col0 + 4] = a1;
    }
    __syncthreads();

    // ---- WMMA phase: wave w owns 16x16 output tile at columns [16w,16w+16) ----
    {
        const int base = wave * 16;
        const int colL = lane & 15;     // N index for B/C, M index for A
        const int half = lane >> 4;     // lane group 0/1
        const int colG = base + colL;   // global output column for B/C

        // C fragment init from gather sums (16x16 f32 C/D layout)
        v8f c;
        #pragma unroll
        for (int v = 0; v < 8; ++v)
            c[v] = s_emb[(v + half * 8) * D + colG];

        // 62(+2 pad) K accumulated as 16 chained f32 WMMAs (K=4 each)
        #pragma unroll
        for (int kc = 0; kc < 16; ++kc) {
            const int kb = kc * 4 + half * 2;  // lanes 0-15: K=0,1 ; lanes 16-31: K=2,3
            v2f a = *(const v2f*)&s_feat[colL * 64 + kb];      // A: 16x4 f32 fragment
            // two adjacent row pointers, forced to the global_load path
            gptr r0 = (gptr)(unsigned long long)s_wrow[kb];
            gptr r1 = (gptr)(unsigned long long)s_wrow[kb + 1];
            v2f b;                                             // B: 4x16 f32 fragment
            b.x = r0[colG];
            b.y = r1[colG];
            c = __builtin_amdgcn_wmma_f32_16x16x4_f32(
                    /*neg_a=*/false, a, /*neg_b=*/false, b,
                    /*c_mod=*/(short)0, c, /*reuse_a=*/false, /*reuse_b=*/false);
        }

        // masked store in C-layout (fast path: whole block in-bounds)
        if (row0 + RPB <= N) {
            #pragma unroll
            for (int v = 0; v < 8; ++v) {
                int r  = v + half * 8;
                int sp = s_ent[r * NCOL];
                float val = (sp != 0 && sp != 1) ? c[v] : 0.0f;
                out[(size_t)(row0 + r) * D + colG] = val;
            }
        } else {
            #pragma unroll
            for (int v = 0; v < 8; ++v) {
                int r  = v + half * 8;
                int gr = row0 + r;
                if (gr < N) {
                    int sp = s_ent[r * NCOL];
                    float val = (sp != 0 && sp != 1) ? c[v] : 0.0f;
                    out[(size_t)gr * D + colG] = val;
                }
            }
        }
    }

    // ---- mask output (float 0/1) at d_out + N*D ----
    if (tid < RPB) {
        int gr = row0 + tid;
        if (gr < N) {
            int sp = s_ent[tid * NCOL];
            out[(size_t)N * D + gr] = (sp != 0 && sp != 1) ? 1.0f : 0.0f;
        }
    }
}

extern "C" void kernel_launch(void* const* d_in, const int* in_sizes, int n_in,
                              void* d_out, int out_size, void* d_ws, size_t ws_size,
                              hipStream_t stream) {
    const int*   entity = (const int*)  d_in[0];
    const float* Wsp = (const float*)d_in[1];
    const float* bsp = (const float*)d_in[2];
    const float* Wab = (const float*)d_in[3];
    const float* bab = (const float*)d_in[4];
    const float* Wit = (const float*)d_in[5];
    const float* bitm= (const float*)d_in[6];
    const float* Wm1 = (const float*)d_in[7];
    const float* Wm2 = (const float*)d_in[8];
    const float* bmv = (const float*)d_in[9];
    const float* Wlv = (const float*)d_in[10];
    const float* blv = (const float*)d_in[11];
    const float* Whp = (const float*)d_in[12];
    const float* bhp = (const float*)d_in[13];
    const float* Wvl = (const float*)d_in[14];
    const float* bvl = (const float*)d_in[15];
    const float* Wft = (const float*)d_in[16];
    const float* bft = (const float*)d_in[17];
    const float* Wbl = (const float*)d_in[18];
    const float* bbl = (const float*)d_in[19];

    const int N = in_sizes[0] / NCOL;
    const int blocks = (N + RPB - 1) / RPB;

    encoder_wmma_kernel<<<blocks, THREADS, 0, stream>>>(
        entity, Wsp, bsp, Wab, bab, Wit, bitm, Wm1, Wm2, bmv,
        Wlv, blv, Whp, bhp, Wvl, bvl, Wft, bft, Wbl, bbl,
        (float*)d_out, N);
}